// GATConv_19370302505610
// MI455X (gfx1250) — compile-verified
//
#include <hip/hip_runtime.h>
#include <math.h>

typedef float v2f __attribute__((ext_vector_type(2)));
typedef float v8f __attribute__((ext_vector_type(8)));

#define N_NODES 100000
#define N_EDGES 1600000
#define F_IN    256
#define F_OUT   64

// ---------------- ordered-int float max helpers ----------------
__device__ __forceinline__ int f32_to_ordered(float f) {
    int i = __float_as_int(f);
    return (i >= 0) ? i : (i ^ 0x7FFFFFFF);
}
__device__ __forceinline__ float ordered_to_f32(int i) {
    return __int_as_float((i >= 0) ? i : (i ^ 0x7FFFFFFF));
}

// ---------------- kernel 1: x = A @ Wv  +  q = x@wq, k = x@wk ----------------
// 256 threads = 8 waves; each wave computes a 16x64 output tile via
// V_WMMA_F32_16X16X4_F32 (4 N-tiles, K stepped by 4). Wv staged in LDS.
__global__ __launch_bounds__(256)
void gat_gemm_qk(const float* __restrict__ A,
                 const float* __restrict__ Wv,
                 const float* __restrict__ wq,
                 const float* __restrict__ wk,
                 float* __restrict__ x,
                 float* __restrict__ qout,
                 float* __restrict__ kout)
{
    __shared__ float sWv[F_IN * F_OUT]; // 64 KB of the 320 KB WGP LDS
    {
        const float4* src = reinterpret_cast<const float4*>(Wv);
        float4*       dst = reinterpret_cast<float4*>(sWv);
        const int total = (F_IN * F_OUT) / 4; // 4096 float4
        for (int i = threadIdx.x; i < total; i += blockDim.x) dst[i] = src[i];
    }
    __syncthreads();

    const int lane    = threadIdx.x & 31;
    const int wave    = threadIdx.x >> 5;
    const int rowBase = blockIdx.x * 128 + wave * 16;
    if (rowBase >= N_NODES) return; // wave-uniform exit (after the only barrier)

    const int halfSel = lane >> 4;   // 0: K even pair, 1: K odd pair (A/B frag layout)
    const int l16     = lane & 15;

    int aRow = rowBase + l16;
    if (aRow >= N_NODES) aRow = N_NODES - 1;         // clamp loads, mask stores later
    const float* aPtr = A + (size_t)aRow * F_IN;

    v8f acc[4];
    acc[0] = v8f{}; acc[1] = v8f{}; acc[2] = v8f{}; acc[3] = v8f{};

    for (int k = 0; k < F_IN; k += 4) {
        const int kk = k + halfSel * 2;
        // A fragment: 16x4 f32 -> 2 VGPRs (K, K+1 per lane), contiguous 8B load
        float2 av = *reinterpret_cast<const float2*>(aPtr + kk);
        v2f afrag; afrag[0] = av.x; afrag[1] = av.y;

        const float* b0 = sWv + (size_t)kk * F_OUT;
        const float* b1 = b0 + F_OUT;
#pragma unroll
        for (int t = 0; t < 4; ++t) {
            v2f bfrag; bfrag[0] = b0[t * 16 + l16]; bfrag[1] = b1[t * 16 + l16];
            acc[t] = __builtin_amdgcn_wmma_f32_16x16x4_f32(
                false, afrag, false, bfrag, (short)0, acc[t], false, false);
        }
    }

    // ---- store x tile: lane half h holds rows rowBase + h*8 + i in VGPR i ----
    const int rbase = rowBase + halfSel * 8;
    if (rowBase + 16 <= N_NODES) {
        // fast path (all blocks except the tail): wave-uniform, no per-element guards
        float* xp = x + (size_t)rbase * F_OUT + l16;
#pragma unroll
        for (int t = 0; t < 4; ++t) {
#pragma unroll
            for (int i = 0; i < 8; ++i) {
                xp[(size_t)i * F_OUT + t * 16] = acc[t][i];
            }
        }
    } else {
#pragma unroll
        for (int t = 0; t < 4; ++t) {
#pragma unroll
            for (int i = 0; i < 8; ++i) {
                const int r = rbase + i;
                if (r < N_NODES) x[(size_t)r * F_OUT + t * 16 + l16] = acc[t][i];
            }
        }
    }

    // ---- fused q/k: per-lane partial dot, then 16-lane butterfly reduce ----
    float wqv[4], wkv[4];
#pragma unroll
    for (int t = 0; t < 4; ++t) { wqv[t] = wq[t * 16 + l16]; wkv[t] = wk[t * 16 + l16]; }

    float pq[8], pk[8];
#pragma unroll
    for (int i = 0; i < 8; ++i) {
        pq[i] = acc[0][i] * wqv[0] + acc[1][i] * wqv[1] + acc[2][i] * wqv[2] + acc[3][i] * wqv[3];
        pk[i] = acc[0][i] * wkv[0] + acc[1][i] * wkv[1] + acc[2][i] * wkv[2] + acc[3][i] * wkv[3];
    }
#pragma unroll
    for (int off = 1; off < 16; off <<= 1) {
#pragma unroll
        for (int i = 0; i < 8; ++i) {
            pq[i] += __shfl_xor(pq[i], off, 32); // stays within each 16-lane half
            pk[i] += __shfl_xor(pk[i], off, 32);
        }
    }
    if (l16 == 0) {
        if (rowBase + 16 <= N_NODES) {
#pragma unroll
            for (int i = 0; i < 8; ++i) { qout[rbase + i] = pq[i]; kout[rbase + i] = pk[i]; }
        } else {
#pragma unroll
            for (int i = 0; i < 8; ++i) {
                if (rbase + i < N_NODES) { qout[rbase + i] = pq[i]; kout[rbase + i] = pk[i]; }
            }
        }
    }
}

// ---------------- kernel 2: init out = b, m = -inf(ordered), denom = 0 ------
__global__ __launch_bounds__(256)
void gat_init(const float* __restrict__ b, float* __restrict__ out,
              int* __restrict__ m_i, float* __restrict__ denom)
{
    const int idx = blockIdx.x * blockDim.x + threadIdx.x;
    if (idx < N_NODES * F_OUT) out[idx] = b[idx & (F_OUT - 1)];
    if (idx < N_NODES) { m_i[idx] = (int)0x80000000; denom[idx] = 0.0f; }
}

// ---------------- kernel 3: edge logits + segment max ------------------------
__global__ __launch_bounds__(256)
void gat_edge_logits(const int* __restrict__ row, const int* __restrict__ col,
                     const float* __restrict__ q, const float* __restrict__ kv,
                     float* __restrict__ attn, int* __restrict__ m_i)
{
    const int e = blockIdx.x * blockDim.x + threadIdx.x;
    if (e >= N_EDGES) return;
    const int r = row[e], c = col[e];
    float a = kv[c] + q[r];
    a = (a > 0.0f) ? a : 0.2f * a;          // leaky_relu, slope 0.2
    attn[e] = a;
    atomicMax(&m_i[r], f32_to_ordered(a));
}

// ---------------- kernel 4: exp(a - m) + segment sum -------------------------
__global__ __launch_bounds__(256)
void gat_edge_exp(const int* __restrict__ row, float* __restrict__ attn,
                  const int* __restrict__ m_i, float* __restrict__ denom)
{
    const int e = blockIdx.x * blockDim.x + threadIdx.x;
    if (e >= N_EDGES) return;
    const int r = row[e];
    const float w = __expf(attn[e] - ordered_to_f32(m_i[r]));
    attn[e] = w;
    atomicAdd(&denom[r], w);
}

// ---------------- kernel 5: normalized attention-weighted aggregation --------
// row is sorted: each 64-thread group (thread = one output feature) walks a
// contiguous edge chunk, accumulating in a register and flushing with a
// single atomicAdd per row run (amortizes atomics by the average degree ~16).
#define EDGES_PER_GROUP 256
__global__ __launch_bounds__(256)
void gat_aggregate(const int* __restrict__ row, const int* __restrict__ col,
                   const float* __restrict__ attn, const float* __restrict__ denom,
                   const float* __restrict__ x, float* __restrict__ out)
{
    const int group   = blockIdx.x * (blockDim.x / F_OUT) + (threadIdx.x / F_OUT);
    const int f       = threadIdx.x & (F_OUT - 1);
    const long long e0 = (long long)group * EDGES_PER_GROUP;
    if (e0 >= N_EDGES) return;
    long long e1 = e0 + EDGES_PER_GROUP;
    if (e1 > N_EDGES) e1 = N_EDGES;

    int   curRow = -1;
    float acc    = 0.0f;
    float dinv   = 0.0f;

    for (long long e = e0; e < e1; ++e) {
        if (e + 8 < N_EDGES)  // hide gather latency: global_prefetch of upcoming row
            __builtin_prefetch(&x[(size_t)col[e + 8] * F_OUT + f], 0, 0);
        const int   r = row[e];
        const int   c = col[e];
        const float w = attn[e];
        if (r != curRow) {
            if (curRow >= 0) atomicAdd(&out[(size_t)curRow * F_OUT + f], acc);
            acc    = 0.0f;
            curRow = r;
            dinv   = 1.0f / denom[r];
        }
        acc += (w * dinv) * x[(size_t)c * F_OUT + f];
    }
    if (curRow >= 0) atomicAdd(&out[(size_t)curRow * F_OUT + f], acc);
}

// ---------------- host launcher ----------------------------------------------
extern "C" void kernel_launch(void* const* d_in, const int* in_sizes, int n_in,
                              void* d_out, int out_size, void* d_ws, size_t ws_size,
                              hipStream_t stream)
{
    const float* A   = (const float*)d_in[0]; // node_features [N,256]
    const int*   row = (const int*)  d_in[1]; // [E], sorted
    const int*   col = (const int*)  d_in[2]; // [E]
    const float* Wv  = (const float*)d_in[3]; // [256,64]
    const float* wq  = (const float*)d_in[4]; // [64]
    const float* wk  = (const float*)d_in[5]; // [64]
    const float* b   = (const float*)d_in[6]; // [64]
    float* out = (float*)d_out;               // [N,64]

    // workspace layout
    char* ws = (char*)d_ws;
    float* x     = (float*)ws; ws += sizeof(float) * (size_t)N_NODES * F_OUT;
    float* q     = (float*)ws; ws += sizeof(float) * N_NODES;
    float* kv    = (float*)ws; ws += sizeof(float) * N_NODES;
    int*   m_i   = (int*)  ws; ws += sizeof(int)   * N_NODES;
    float* denom = (float*)ws; ws += sizeof(float) * N_NODES;
    float* attn  = (float*)ws; ws += sizeof(float) * (size_t)N_EDGES;

    // 1) GEMM + q/k (WMMA f32 16x16x4)
    gat_gemm_qk<<<(N_NODES + 127) / 128, 256, 0, stream>>>(A, Wv, wq, wk, x, q, kv);
    // 2) init out/m/denom
    gat_init<<<((N_NODES * F_OUT) + 255) / 256, 256, 0, stream>>>(b, out, m_i, denom);
    // 3) logits + segment max
    gat_edge_logits<<<(N_EDGES + 255) / 256, 256, 0, stream>>>(row, col, q, kv, attn, m_i);
    // 4) exp + segment sum
    gat_edge_exp<<<(N_EDGES + 255) / 256, 256, 0, stream>>>(row, attn, m_i, denom);
    // 5) aggregation
    {
        const int nGroups = (N_EDGES + EDGES_PER_GROUP - 1) / EDGES_PER_GROUP;
        const int groupsPerBlock = 256 / F_OUT;
        const int nBlocks = (nGroups + groupsPerBlock - 1) / groupsPerBlock;
        gat_aggregate<<<nBlocks, 256, 0, stream>>>(row, col, attn, denom, x, out);
    }
}